// MagnitudePruningCallback_46514495816081
// MI455X (gfx1250) — compile-verified
//
#include <hip/hip_runtime.h>
#include <stdint.h>

// MagnitudePruning on MI455X (gfx1250).
// Memory-bound radix select (11/11/10 bits) over monotone-mapped float bits,
// then masked copy. With big-enough d_ws, pass 0 caches mapped mag bits so
// later passes stream 180MB instead of 360MB: ~1.44 GB total -> ~62us at
// 23.3 TB/s. WMMA does not apply (no matrix structure). CDNA5 paths:
// global_load/store_b128, LDS ds_add atomics, global_prefetch_b8.

namespace {

constexpr uint32_t kN    = 45088768u;   // 4096 * 11008
constexpr uint32_t kNVec = kN / 4u;     // exact: 11,272,192 x (128-bit)
constexpr int kHistOff    = 8;          // ws[0]=prefix ws[1]=mask ws[2]=rank
constexpr int kMaxBuckets = 2048;
constexpr uint32_t kStateWords = 4096;  // 16 KB state+hist region at ws start

// Opaque data barrier: prevents mul+add fusion into FMA (the reference / XLA
// performs separately-rounded f32 ops; contraction would change the bits).
__device__ __forceinline__ float fence_f(float v) {
  asm("" : "+v"(v));
  return v;
}

// mag = (t*m + |x|)/(t+1), each op rounded separately, then monotone bit map:
// unsigned order of mapped bits == total float order.
__device__ __forceinline__ uint32_t mag_mapped(float xv, float mv, float tf, float tp1) {
  float prod = fence_f(__fmul_rn(tf, mv));          // no FMA contraction
  float num  = __fadd_rn(prod, __builtin_fabsf(xv));
  float mag  = __fdiv_rn(num, tp1);
  uint32_t b = __float_as_uint(mag);
  return b ^ (uint32_t(int32_t(b) >> 31) | 0x80000000u);
}

__global__ __launch_bounds__(256) void init_kernel(const float* __restrict__ sparsity,
                                                   uint32_t* __restrict__ ws) {
  if (threadIdx.x == 0) {
    // idx = clip(floor(sparsity*n - 1), 0, n-1), f32 mul and sub rounded
    // separately exactly like the reference.
    float prod = fence_f(__fmul_rn(sparsity[0], 45088768.0f));
    float f = floorf(__fsub_rn(prod, 1.0f));
    long long idx = (long long)f;
    if (idx < 0) idx = 0;
    if (idx > (long long)(kN - 1u)) idx = (long long)(kN - 1u);
    ws[0] = 0u;              // selected prefix (mapped bits)
    ws[1] = 0u;              // prefix validity mask
    ws[2] = (uint32_t)idx;   // remaining 0-based rank
  }
  for (int i = (int)threadIdx.x; i < kMaxBuckets; i += (int)blockDim.x)
    ws[kHistOff + i] = 0u;
}

// ---------------- Fallback path: recompute mag bits every pass ----------------

__global__ __launch_bounds__(256) void hist_kernel(const float4* __restrict__ x4,
                                                   const float4* __restrict__ m4,
                                                   const int* __restrict__ tptr,
                                                   uint32_t* __restrict__ ws,
                                                   uint32_t shift, uint32_t nbuckets) {
  __shared__ uint32_t lh[kMaxBuckets];
  for (uint32_t i = threadIdx.x; i < nbuckets; i += blockDim.x) lh[i] = 0u;
  __syncthreads();

  const uint32_t prefix = ws[0];
  const uint32_t pmask  = ws[1];
  const uint32_t bmask  = nbuckets - 1u;
  const float tf  = (float)tptr[0];
  const float tp1 = __fadd_rn(tf, 1.0f);

  const uint32_t stride = gridDim.x * blockDim.x;
  for (uint32_t i = blockIdx.x * blockDim.x + threadIdx.x; i < kNVec; i += stride) {
    if (i + stride < kNVec) {
      __builtin_prefetch(&x4[i + stride], 0, 0);
      __builtin_prefetch(&m4[i + stride], 0, 0);
    }
    const float4 xv = x4[i];
    const float4 mv = m4[i];
    const uint32_t b0 = mag_mapped(xv.x, mv.x, tf, tp1);
    const uint32_t b1 = mag_mapped(xv.y, mv.y, tf, tp1);
    const uint32_t b2 = mag_mapped(xv.z, mv.z, tf, tp1);
    const uint32_t b3 = mag_mapped(xv.w, mv.w, tf, tp1);
    if ((b0 & pmask) == prefix) atomicAdd(&lh[(b0 >> shift) & bmask], 1u);
    if ((b1 & pmask) == prefix) atomicAdd(&lh[(b1 >> shift) & bmask], 1u);
    if ((b2 & pmask) == prefix) atomicAdd(&lh[(b2 >> shift) & bmask], 1u);
    if ((b3 & pmask) == prefix) atomicAdd(&lh[(b3 >> shift) & bmask], 1u);
  }
  __syncthreads();
  for (uint32_t i = threadIdx.x; i < nbuckets; i += blockDim.x) {
    const uint32_t c = lh[i];
    if (c) atomicAdd(&ws[kHistOff + i], c);
  }
}

__global__ __launch_bounds__(256) void apply_kernel(const float4* __restrict__ x4,
                                                    const float4* __restrict__ m4,
                                                    const int* __restrict__ tptr,
                                                    const uint32_t* __restrict__ ws,
                                                    float4* __restrict__ out4) {
  const uint32_t thr = ws[0];
  const float tf  = (float)tptr[0];
  const float tp1 = __fadd_rn(tf, 1.0f);

  const uint32_t stride = gridDim.x * blockDim.x;
  for (uint32_t i = blockIdx.x * blockDim.x + threadIdx.x; i < kNVec; i += stride) {
    if (i + stride < kNVec) {
      __builtin_prefetch(&x4[i + stride], 0, 0);
      __builtin_prefetch(&m4[i + stride], 0, 0);
    }
    const float4 xv = x4[i];
    const float4 mv = m4[i];
    float4 r;
    r.x = __fmul_rn(xv.x, (mag_mapped(xv.x, mv.x, tf, tp1) >= thr) ? 1.0f : 0.0f);
    r.y = __fmul_rn(xv.y, (mag_mapped(xv.y, mv.y, tf, tp1) >= thr) ? 1.0f : 0.0f);
    r.z = __fmul_rn(xv.z, (mag_mapped(xv.z, mv.z, tf, tp1) >= thr) ? 1.0f : 0.0f);
    r.w = __fmul_rn(xv.w, (mag_mapped(xv.w, mv.w, tf, tp1) >= thr) ? 1.0f : 0.0f);
    out4[i] = r;
  }
}

// --------------- Cached path: pass 0 stores mapped bits in d_ws ---------------

__global__ __launch_bounds__(256) void hist_store_kernel(const float4* __restrict__ x4,
                                                         const float4* __restrict__ m4,
                                                         const int* __restrict__ tptr,
                                                         uint32_t* __restrict__ ws,
                                                         uint4* __restrict__ bits4) {
  __shared__ uint32_t lh[kMaxBuckets];
  for (uint32_t i = threadIdx.x; i < (uint32_t)kMaxBuckets; i += blockDim.x) lh[i] = 0u;
  __syncthreads();

  const float tf  = (float)tptr[0];
  const float tp1 = __fadd_rn(tf, 1.0f);

  const uint32_t stride = gridDim.x * blockDim.x;
  for (uint32_t i = blockIdx.x * blockDim.x + threadIdx.x; i < kNVec; i += stride) {
    if (i + stride < kNVec) {
      __builtin_prefetch(&x4[i + stride], 0, 0);
      __builtin_prefetch(&m4[i + stride], 0, 0);
    }
    const float4 xv = x4[i];
    const float4 mv = m4[i];
    uint4 b;
    b.x = mag_mapped(xv.x, mv.x, tf, tp1);
    b.y = mag_mapped(xv.y, mv.y, tf, tp1);
    b.z = mag_mapped(xv.z, mv.z, tf, tp1);
    b.w = mag_mapped(xv.w, mv.w, tf, tp1);
    bits4[i] = b;                       // cache mapped bits for later passes
    atomicAdd(&lh[b.x >> 21], 1u);      // pass 0: top 11 bits, no filter
    atomicAdd(&lh[b.y >> 21], 1u);
    atomicAdd(&lh[b.z >> 21], 1u);
    atomicAdd(&lh[b.w >> 21], 1u);
  }
  __syncthreads();
  for (uint32_t i = threadIdx.x; i < (uint32_t)kMaxBuckets; i += blockDim.x) {
    const uint32_t c = lh[i];
    if (c) atomicAdd(&ws[kHistOff + i], c);
  }
}

__global__ __launch_bounds__(256) void hist_cached_kernel(const uint4* __restrict__ bits4,
                                                          uint32_t* __restrict__ ws,
                                                          uint32_t shift, uint32_t nbuckets) {
  __shared__ uint32_t lh[kMaxBuckets];
  for (uint32_t i = threadIdx.x; i < nbuckets; i += blockDim.x) lh[i] = 0u;
  __syncthreads();

  const uint32_t prefix = ws[0];
  const uint32_t pmask  = ws[1];
  const uint32_t bmask  = nbuckets - 1u;

  const uint32_t stride = gridDim.x * blockDim.x;
  for (uint32_t i = blockIdx.x * blockDim.x + threadIdx.x; i < kNVec; i += stride) {
    if (i + stride < kNVec) __builtin_prefetch(&bits4[i + stride], 0, 0);
    const uint4 b = bits4[i];
    if ((b.x & pmask) == prefix) atomicAdd(&lh[(b.x >> shift) & bmask], 1u);
    if ((b.y & pmask) == prefix) atomicAdd(&lh[(b.y >> shift) & bmask], 1u);
    if ((b.z & pmask) == prefix) atomicAdd(&lh[(b.z >> shift) & bmask], 1u);
    if ((b.w & pmask) == prefix) atomicAdd(&lh[(b.w >> shift) & bmask], 1u);
  }
  __syncthreads();
  for (uint32_t i = threadIdx.x; i < nbuckets; i += blockDim.x) {
    const uint32_t c = lh[i];
    if (c) atomicAdd(&ws[kHistOff + i], c);
  }
}

__global__ __launch_bounds__(256) void apply_cached_kernel(const float4* __restrict__ x4,
                                                           const uint4* __restrict__ bits4,
                                                           const uint32_t* __restrict__ ws,
                                                           float4* __restrict__ out4) {
  const uint32_t thr = ws[0];
  const uint32_t stride = gridDim.x * blockDim.x;
  for (uint32_t i = blockIdx.x * blockDim.x + threadIdx.x; i < kNVec; i += stride) {
    if (i + stride < kNVec) {
      __builtin_prefetch(&x4[i + stride], 0, 0);
      __builtin_prefetch(&bits4[i + stride], 0, 0);
    }
    const float4 xv = x4[i];
    const uint4  b  = bits4[i];
    float4 r;
    r.x = __fmul_rn(xv.x, (b.x >= thr) ? 1.0f : 0.0f);
    r.y = __fmul_rn(xv.y, (b.y >= thr) ? 1.0f : 0.0f);
    r.z = __fmul_rn(xv.z, (b.z >= thr) ? 1.0f : 0.0f);
    r.w = __fmul_rn(xv.w, (b.w >= thr) ? 1.0f : 0.0f);
    out4[i] = r;
  }
}

// --------------------------------- Scan ---------------------------------

__global__ __launch_bounds__(256) void scan_kernel(uint32_t* __restrict__ ws,
                                                   uint32_t shift, uint32_t nbuckets) {
  if (threadIdx.x == 0) {
    const uint32_t k = ws[2];
    uint32_t cumBefore = 0u;
    uint32_t sel = nbuckets - 1u;
    uint32_t cum = 0u;
    bool found = false;
    for (uint32_t b = 0; b < nbuckets; ++b) {
      const uint32_t c = ws[kHistOff + b];
      if (cum + c > k) { sel = b; cumBefore = cum; found = true; break; }
      cum += c;
    }
    if (!found) { sel = nbuckets - 1u; cumBefore = (cum > k) ? k : cum; }
    ws[0] |= (sel << shift);
    ws[1] |= ((nbuckets - 1u) << shift);
    ws[2] = k - cumBefore;
  }
  __syncthreads();
  for (uint32_t i = threadIdx.x; i < (uint32_t)kMaxBuckets; i += blockDim.x)
    ws[kHistOff + i] = 0u;
}

}  // namespace

extern "C" void kernel_launch(void* const* d_in, const int* in_sizes, int n_in,
                              void* d_out, int out_size, void* d_ws, size_t ws_size,
                              hipStream_t stream) {
  (void)in_sizes; (void)n_in; (void)out_size;

  const float4* x4 = (const float4*)d_in[0];       // x         f32 [4096,11008]
  const float4* m4 = (const float4*)d_in[1];       // magnitude f32 [4096,11008]
  const float*  sp = (const float*)d_in[2];        // sparsity  f32 [1]
  /* d_in[3] = mask (bool) is unused by the reference computation */
  const int*    tp = (const int*)d_in[4];          // t int [1]
  float4* out4 = (float4*)d_out;

  uint32_t* ws = (uint32_t*)d_ws;
  uint32_t* magbits = ws + kStateWords;            // 16 KB into ws
  const bool cached =
      ws_size >= (size_t)kStateWords * 4u + (size_t)kN * 4u;

  const dim3 blk(256);
  const dim3 grid(2048);

  init_kernel<<<1, blk, 0, stream>>>(sp, ws);

  if (cached) {
    // Pass 0 computes + caches mapped bits; passes 1-2 stream the cache.
    hist_store_kernel<<<grid, blk, 0, stream>>>(x4, m4, tp, ws, (uint4*)magbits);
    scan_kernel<<<1, blk, 0, stream>>>(ws, 21u, 2048u);

    hist_cached_kernel<<<grid, blk, 0, stream>>>((const uint4*)magbits, ws, 10u, 2048u);
    scan_kernel<<<1, blk, 0, stream>>>(ws, 10u, 2048u);

    hist_cached_kernel<<<grid, blk, 0, stream>>>((const uint4*)magbits, ws, 0u, 1024u);
    scan_kernel<<<1, blk, 0, stream>>>(ws, 0u, 1024u);

    apply_cached_kernel<<<grid, blk, 0, stream>>>(x4, (const uint4*)magbits, ws, out4);
  } else {
    hist_kernel<<<grid, blk, 0, stream>>>(x4, m4, tp, ws, 21u, 2048u);
    scan_kernel<<<1, blk, 0, stream>>>(ws, 21u, 2048u);

    hist_kernel<<<grid, blk, 0, stream>>>(x4, m4, tp, ws, 10u, 2048u);
    scan_kernel<<<1, blk, 0, stream>>>(ws, 10u, 2048u);

    hist_kernel<<<grid, blk, 0, stream>>>(x4, m4, tp, ws, 0u, 1024u);
    scan_kernel<<<1, blk, 0, stream>>>(ws, 0u, 1024u);

    apply_kernel<<<grid, blk, 0, stream>>>(x4, m4, tp, ws, out4);
  }
}